// Bottleneck_refine_43533788512663
// MI455X (gfx1250) — compile-verified
//
#include <hip/hip_runtime.h>
#include <hip/hip_bf16.h>

// ---------------------------------------------------------------------------
// Fused masked grouped bottleneck (conv1x1 -> conv3x3 -> conv1x1 + residual)
// for MI455X / gfx1250. One workgroup per (group, 16x16 patch); all
// intermediates live in LDS; matrix math via V_WMMA_F32_16X16X4_F32 (wave32);
// conv2 weights staged into LDS by the Tensor Data Mover (overlapped with
// conv1 compute), with TDM LDS-padding used to kill bank conflicts.
// ---------------------------------------------------------------------------

typedef __attribute__((ext_vector_type(2))) float v2f;
typedef __attribute__((ext_vector_type(8))) float v8f;
typedef __attribute__((ext_vector_type(4))) unsigned int v4u;
typedef __attribute__((ext_vector_type(8))) int v8i;
typedef __attribute__((ext_vector_type(4))) int v4i;

#define NGROUP 4
#define CIN    256        // input channels per group (conv1 K)
#define CMID   64         // mid channels per group
#define COUT   256        // output channels per group
#define HW     128
#define NPIX   (HW * HW)  // 16384
#define HALO   18
#define NHALO  (HALO * HALO)  // 324
#define NT_H   21             // ceil(324/16) halo pixel tiles
#define NP     (NT_H * 16)    // 336, padded halo row stride (floats)
#define SUS    264            // 256+8 pad: de-conflicts lane-half rows
#define W2DW   (CMID * CMID * 9)       // 36864 dwords of group-g conv2 weights
#define W2PAD  (W2DW + (W2DW / 256))   // +1 dword per 256 (TDM pad) = 37008

// D = A(16x4) * B(4x16) + C, fp32, wave32.
// A layout: lane l<16 -> row M=l, K={0,1}; lane l>=16 -> row M=l-16, K={2,3}
// B layout: vgpr0 -> K=0 (lanes 0-15) / K=2 (lanes 16-31); vgpr1 -> K=1 / K=3
// C layout: vgpr r -> M=r (lanes 0-15) / M=r+8 (lanes 16-31), N=lane&15
__device__ __forceinline__ v8f wmma4(v2f a, v2f b, v8f c) {
  return __builtin_amdgcn_wmma_f32_16x16x4_f32(
      /*neg_a=*/false, a, /*neg_b=*/false, b,
      /*c_mod=*/(short)0, c, /*reuse_a=*/false, /*reuse_b=*/false);
}

__global__ __launch_bounds__(256)
void bottleneck_refine_fused(const float* __restrict__ x,
                             const float* __restrict__ mask,
                             const float* __restrict__ w1,
                             const float* __restrict__ w2,
                             const float* __restrict__ w3,
                             float* __restrict__ out) {
  extern __shared__ float smem[];
  float* sT  = smem;                        // [CMID][NP]  relu(conv1) halo
  float* sU  = smem + CMID * NP;            // [CMID][SUS] relu(conv2) interior
  float* sW2 = smem + CMID * NP + CMID * SUS;  // [W2PAD] TDM-staged conv2 wts

  const int blk   = blockIdx.x;             // 256 blocks: (group, patch)
  const int g     = blk >> 6;
  const int patch = blk & 63;
  const int py0   = (patch >> 3) * 16;
  const int px0   = (patch & 7) * 16;

  const int tid  = threadIdx.x;
  const int lane = tid & 31;
  const int wid  = tid >> 5;                // 8 wave32s
  const int ln   = lane & 15;               // N / M index within fragment
  const int khi  = lane >> 4;               // lane-half: K offset {0,2}

  const float* maskg = mask + g * 64;
  const bool on = maskg[patch] != 0.0f;

  const float* xg  = x   + (size_t)g * CIN * NPIX;   // group in/residual chans
  float*      outg = out + (size_t)g * CIN * NPIX;

  if (!on) {
    // Masked-off patch: out = relu(x) on this group's 256 channels, 16x16 px.
    const int nUnits = COUT * 16 * 4;                // 16384 float4 units
    for (int u = tid; u < nUnits; u += 256) {
      const int ch  = u >> 6;
      const int rem = u & 63;
      const size_t off = (size_t)ch * NPIX + (size_t)(py0 + (rem >> 2)) * HW +
                         px0 + (rem & 3) * 4;
      float4 v = *(const float4*)(xg + off);
      v.x = fmaxf(v.x, 0.f); v.y = fmaxf(v.y, 0.f);
      v.z = fmaxf(v.z, 0.f); v.w = fmaxf(v.w, 0.f);
      *(float4*)(outg + off) = v;
    }
    return;
  }

  // --- TDM: DMA this group's conv2 weights (147KB) into LDS, overlapped
  // with the whole conv1 stage. 1-D tile of 36864 dwords; LDS padding of
  // 1 dword per 256 breaks the 576-dword row stride's bank alignment.
  if (wid == 0) {
    const unsigned long long ga =
        (unsigned long long)(const void*)(w2 + (size_t)g * W2DW);
    const unsigned int ldsOff = (unsigned int)(unsigned long long)(void*)sW2;
    v4u g0 = {1u,                                  // count=1 user descriptor
              ldsOff,                              // lds_addr
              (unsigned int)ga,                    // global_addr[31:0]
              ((unsigned int)(ga >> 32) & 0x01FFFFFFu) | 0x80000000u};  // +type=2
    v8i g1 = {(int)0x01D20000u,  // data_size=4B | pad_enable | pad_int=256dw | pad_amt=1dw
              (int)0x90000000u,  // tensor_dim0[15:0]=36864 (bits 63:48)
              (int)0x00010000u,  // tensor_dim0[31:16]=0, tensor_dim1=1
              (int)0x90000000u,  // tile_dim0=36864 (bits 127:112)
              0,                 // tile_dim1=0, tile_dim2=0 (1-D)
              (int)W2DW,         // tensor_dim0_stride[31:0]
              0, 0};
    v4i gz4 = {0, 0, 0, 0};
    v8i gz8 = {0, 0, 0, 0, 0, 0, 0, 0};
    __builtin_amdgcn_tensor_load_to_lds(g0, g1, gz4, gz4, gz8, 0);
  }

  // Warm conv1/conv3 weights (64KB each) into cache while conv1 starts.
  {
    const char* pw1 = (const char*)(w1 + (size_t)g * CMID * CIN);
    const char* pw3 = (const char*)(w3 + (size_t)g * COUT * CMID);
    for (int o = tid * 128; o < CMID * CIN * 4; o += 256 * 128) {
      __builtin_prefetch(pw1 + o, 0, 1);
      __builtin_prefetch(pw3 + o, 0, 1);
    }
  }

  const v8f vzero = {0.f, 0.f, 0.f, 0.f, 0.f, 0.f, 0.f, 0.f};

  // ------------------------------------------------------------------
  // Stage 1: conv1 (1x1, K=256) over 18x18 halo -> sT = mask * relu(.)
  //   M=64 mid channels (4 tiles), N=324 halo px (21 tiles), flat RR.
  // ------------------------------------------------------------------
  for (int t = wid; t < 4 * NT_H; t += 8) {
    const int mt = t & 3;
    const int nt = t >> 2;

    const int hp = nt * 16 + ln;               // linear halo pixel
    const int hy = hp / HALO, hx = hp - hy * HALO;
    const int iy = py0 - 1 + hy;
    const int ix = px0 - 1 + hx;
    bool ok = (hp < NHALO) & (iy >= 0) & (iy < HW) & (ix >= 0) & (ix < HW);
    if (ok) ok = maskg[(iy >> 4) * 8 + (ix >> 4)] != 0.0f;  // neighbor mask
    const int pixOff = ok ? (iy * HW + ix) : 0;             // safe address

    const float* w1row = w1 + (size_t)(g * CMID + mt * 16 + ln) * CIN + 2 * khi;
    const float* xb    = xg + (size_t)(2 * khi) * NPIX + pixOff;

    v8f acc = vzero;
    for (int k = 0; k < CIN; k += 4) {
      v2f a = *(const v2f*)(w1row + k);                      // 8B aligned
      v2f b;
      b.x = xb[(size_t)k * NPIX];
      b.y = xb[(size_t)(k + 1) * NPIX];
      acc = wmma4(a, b, acc);
    }
    float* dst = sT + nt * 16 + ln;
#pragma unroll
    for (int r = 0; r < 8; ++r) {
      const int m = mt * 16 + r + 8 * khi;
      dst[m * NP] = ok ? fmaxf(acc[r], 0.f) : 0.f;  // mask+border+relu folded
    }
  }

  __builtin_amdgcn_s_wait_tensorcnt(0);  // wave0: W2 DMA done (others: no-op)
  __syncthreads();

  // ------------------------------------------------------------------
  // Stage 2: conv2 (3x3, K=64 per tap), all-LDS -> sU = relu(.)
  //   M=64 (4 tiles), N=256 interior px (16 tiles, 1 tile = 1 spatial row).
  //   Each wave: fixed mt, 8 N-tiles register-blocked (A frag reused 8x).
  //   A fragments from TDM-staged sW2 (padded index: d + (d>>8)).
  // ------------------------------------------------------------------
  {
    const int mt     = wid & 3;
    const int ntBase = (wid >> 2) * 8;       // 0 or 8
    v8f acc[8];
#pragma unroll
    for (int j = 0; j < 8; ++j) acc[j] = vzero;

    const int aBase = (mt * 16 + ln) * (CMID * 9) + 18 * khi;  // data dword idx

    for (int dy = 0; dy < 3; ++dy) {
      for (int dx = 0; dx < 3; ++dx) {
        const int tapOff = aBase + dy * 3 + dx;
        const float* tb  = sT + (size_t)(2 * khi) * NP + dy * HALO + dx + ln;
        for (int k = 0; k < CMID; k += 4) {
          const int d0 = tapOff + 9 * k;     // i = k + 2*khi
          const int d1 = d0 + 9;             // i = k + 2*khi + 1
          v2f a;
          a.x = sW2[d0 + (d0 >> 8)];
          a.y = sW2[d1 + (d1 >> 8)];
          const float* tk = tb + (size_t)k * NP;
#pragma unroll
          for (int j = 0; j < 8; ++j) {
            const int nt = ntBase + j;       // output spatial row
            v2f b;
            b.x = tk[(nt + dy) * HALO];      // sT[k+2khi][(nt+2*dy)*18+ln+dx]
            b.y = tk[NP + (nt + dy) * HALO];
            acc[j] = wmma4(a, b, acc[j]);
          }
        }
      }
    }
#pragma unroll
    for (int j = 0; j < 8; ++j) {
      float* dst = sU + (ntBase + j) * 16 + ln;
#pragma unroll
      for (int r = 0; r < 8; ++r) {
        const int m = mt * 16 + r + 8 * khi;
        dst[m * SUS] = fmaxf(acc[j][r], 0.f);  // interior mask == 1 here
      }
    }
  }
  __syncthreads();

  // ------------------------------------------------------------------
  // Stage 3: conv3 (1x1, K=64) + residual + relu -> out
  //   M=256 out channels (16 tiles), N=256 px (16 tiles), flat RR.
  // ------------------------------------------------------------------
  for (int t = wid; t < 256; t += 8) {
    const int mt = t & 15;
    const int nt = t >> 4;                   // spatial row

    const float* w3row =
        w3 + (size_t)(g * COUT + mt * 16 + ln) * CMID + 2 * khi;
    const float* ub = sU + (size_t)(2 * khi) * SUS + nt * 16 + ln;

    v8f acc = vzero;
    for (int k = 0; k < CMID; k += 4) {
      v2f a = *(const v2f*)(w3row + k);
      v2f b;
      b.x = ub[(size_t)k * SUS];
      b.y = ub[(size_t)(k + 1) * SUS];
      acc = wmma4(a, b, acc);
    }

    const size_t poff = (size_t)(py0 + nt) * HW + px0 + ln;
#pragma unroll
    for (int r = 0; r < 8; ++r) {
      const int ch = mt * 16 + r + 8 * khi;
      const size_t o = (size_t)ch * NPIX + poff;
      outg[o] = fmaxf(xg[o] + acc[r], 0.f);
    }
  }
}

extern "C" void kernel_launch(void* const* d_in, const int* in_sizes, int n_in,
                              void* d_out, int out_size, void* d_ws,
                              size_t ws_size, hipStream_t stream) {
  const float* x    = (const float*)d_in[0];
  const float* mask = (const float*)d_in[1];
  const float* w1   = (const float*)d_in[2];
  const float* w2   = (const float*)d_in[3];
  const float* w3   = (const float*)d_in[4];
  float* out        = (float*)d_out;

  const size_t shmem =
      (size_t)(CMID * NP + CMID * SUS + W2PAD) * sizeof(float);  // ~295KB
  bottleneck_refine_fused<<<dim3(256), dim3(256), shmem, stream>>>(
      x, mask, w1, w2, w3, out);
}